// MaskRCNN_89326729822373
// MI455X (gfx1250) — compile-verified
//
#include <hip/hip_runtime.h>
#include <hip/hip_bf16.h>

typedef __attribute__((ext_vector_type(2))) float v2f;
typedef __attribute__((ext_vector_type(4))) float f4;
typedef __attribute__((ext_vector_type(8))) float v8f;

#define B_    2
#define P_    128
#define R_    8
#define C_    81
#define HW_   (384 * 384)
#define NNEG  768
#define NPOS  256
#define S_    1024

// float offsets into d_out (concatenated outputs, return order)
#define ROIS_OFF 0
#define CLS_OFF  (B_ * S_ * 4)              // 8192
#define BBOX_OFF (CLS_OFF + B_ * S_ * C_)   // 174080
#define MASK_OFF (BBOX_OFF + B_ * S_ * 4)   // 182272

// ---------------------------------------------------------------------------
// Kernel 1: pos[b,p,r] = (iou >= 0.5) as float, preserving reference quirks.
// 2048 elements total — trivial.
// ---------------------------------------------------------------------------
__global__ void iou_pos_kernel(const float* __restrict__ prop,
                               const float* __restrict__ gtb,
                               float* __restrict__ posf) {
    int g = blockIdx.x * blockDim.x + threadIdx.x;
    if (g >= B_ * P_ * R_) return;
    int b   = g / (P_ * R_);
    int rem = g - b * (P_ * R_);
    int p   = rem >> 3;
    int r   = rem & 7;

    f4 pb = *(const f4*)(prop + (size_t)(b * P_ + p) * 4);
    f4 gb = *(const f4*)(gtb  + (size_t)(b * R_ + r) * 4);

    float a1 = (pb.z - pb.x + 1.0f) * (pb.w - pb.y);   // +1 on width only (quirk)
    float a2 = (gb.z - gb.x + 1.0f) * (gb.w - gb.y);
    float ix1 = fmaxf(pb.x, gb.y);                     // source bug kept: p_x1 vs g_y1
    float iy1 = fmaxf(pb.y, gb.y);
    float ix2 = fminf(pb.z, gb.z);
    float iy2 = fminf(pb.w, gb.w);
    float iw = fmaxf(0.0f, ix2 - ix1 + 1.0f);
    float ih = fmaxf(0.0f, iy2 - iy1 + 1.0f);
    float inter = iw * ih;
    float iou = inter / (a1 + a2 + inter);             // denom incl. inter (quirk)
    posf[g] = (iou >= 0.5f) ? 1.0f : 0.0f;
}

// ---------------------------------------------------------------------------
// Kernel 2: rois + bbox_targets (4-float gated gathers). One thread / sample.
// ---------------------------------------------------------------------------
__global__ void rois_bbox_kernel(const float* __restrict__ prop,
                                 const float* __restrict__ gtb,
                                 const int* __restrict__ negI,
                                 const int* __restrict__ posI,
                                 const float* __restrict__ posf,
                                 float* __restrict__ out) {
    int g = blockIdx.x * blockDim.x + threadIdx.x;
    if (g >= B_ * S_) return;
    int b = g >> 10;
    int s = g & (S_ - 1);
    int idx = (s < NNEG) ? negI[s] : posI[s - NNEG];
    int j = idx >> 3;
    int k = idx & 7;
    float pf   = posf[b * (P_ * R_) + j * R_ + k];
    float gate = (s < NNEG) ? (1.0f - pf) : pf;

    f4 pv = *(const f4*)(prop + (size_t)(b * P_ + j) * 4);
    f4 bv = *(const f4*)(gtb  + (size_t)(b * R_ + k) * 4);
    *(f4*)(out + ROIS_OFF + (size_t)g * 4) = pv * gate;
    *(f4*)(out + BBOX_OFF + (size_t)g * 4) = bv * gate;
}

// ---------------------------------------------------------------------------
// Kernel 3: cls_targets via one-hot WMMA.
//   out[b, 16m:16m+16, :] = A(16x8 one-hot*gate) x G(8x81 gt_classes)
// K=8 -> two chained V_WMMA_F32_16X16X4_F32. Exact in f32 (single nonzero
// per dot product -> no rounding). One wave per M-tile; 128 waves total.
// A layout (ISA 7.12.2, 32-bit 16x4): lanes 0-15 hold K{0,1}, lanes 16-31 K{2,3}.
// B layout (4x16, rows striped across lanes): VGPR0 = K0/K2, VGPR1 = K1/K3.
// D layout: VGPR i -> row i (lanes 0-15) / row i+8 (lanes 16-31), N = lane%16.
// ---------------------------------------------------------------------------
__global__ void cls_wmma_kernel(const float* __restrict__ gcls,
                                const int* __restrict__ negI,
                                const int* __restrict__ posI,
                                const float* __restrict__ posf,
                                float* __restrict__ out) {
    int lane = threadIdx.x;            // 0..31, one wave per block
    int tile = blockIdx.x & 63;        // M-tile (64 tiles of 16 samples)
    int b    = blockIdx.x >> 6;
    int M    = lane & 15;

    // Build one-hot A fragment: row M has gate at column k_s.
    int s   = tile * 16 + M;
    int idx = (s < NNEG) ? negI[s] : posI[s - NNEG];
    int j = idx >> 3;
    int k = idx & 7;
    float pf   = posf[b * (P_ * R_) + j * R_ + k];
    float gate = (s < NNEG) ? (1.0f - pf) : pf;

    int cb = (lane < 16) ? 0 : 2;      // K columns held by this half-wave
    v2f alo, ahi;
    alo.x = (k == cb)     ? gate : 0.0f;   // K = cb
    alo.y = (k == cb + 1) ? gate : 0.0f;   // K = cb+1
    ahi.x = (k == cb + 4) ? gate : 0.0f;   // second WMMA: K+4
    ahi.y = (k == cb + 5) ? gate : 0.0f;

    const float* G = gcls + (size_t)b * (R_ * C_);
    int mbase = (lane < 16) ? 0 : 8;

    for (int n = 0; n < 6; ++n) {      // 6 N-tiles cover 81 (pad to 96)
        int N = n * 16 + M;
        v2f blo = {0.0f, 0.0f}, bhi = {0.0f, 0.0f};
        if (N < C_) {
            blo.x = G[(cb)     * C_ + N];
            blo.y = G[(cb + 1) * C_ + N];
            bhi.x = G[(cb + 4) * C_ + N];
            bhi.y = G[(cb + 5) * C_ + N];
        }
        v8f acc = {};
        acc = __builtin_amdgcn_wmma_f32_16x16x4_f32(
            false, alo, false, blo, (short)0, acc, false, false);
        acc = __builtin_amdgcn_wmma_f32_16x16x4_f32(
            false, ahi, false, bhi, (short)0, acc, false, false);

        if (N < C_) {
            #pragma unroll
            for (int i = 0; i < 8; ++i) {
                int row = tile * 16 + mbase + i;
                out[CLS_OFF + ((size_t)b * S_ + row) * C_ + N] = acc[i];
            }
        }
    }
}

// ---------------------------------------------------------------------------
// Kernel 4: mask_targets — the 302 MB streaming broadcast (the whole runtime).
// dst[b,t,:,:] = gt_masks[b, k_pos[t]] * gate. b128 loads (src set = 9.4 MB,
// stays cached), non-temporal b128 stores (write-once output). Every element
// is written (gate==0 -> zeros), so the poisoned output is fully initialized.
// Grid: 36 blocks x (B*NPOS=512) planes, 256 thr, 4 float4 per thread.
// ---------------------------------------------------------------------------
__global__ void mask_kernel(const float* __restrict__ gmask,
                            const int* __restrict__ posI,
                            const float* __restrict__ posf,
                            float* __restrict__ out) {
    int plane = blockIdx.y;            // 0..511  (= b*NPOS + t)
    int b = plane >> 8;
    int t = plane & (NPOS - 1);
    int idx = posI[t];
    int j = idx >> 3;
    int k = idx & 7;
    float gate = posf[b * (P_ * R_) + j * R_ + k];

    const f4* __restrict__ src = (const f4*)(gmask + (size_t)(b * R_ + k) * HW_);
    f4* __restrict__ dst = (f4*)(out + MASK_OFF + (size_t)plane * HW_);

    int base = blockIdx.x * blockDim.x + threadIdx.x;  // 0..9215
    const int stride = 36 * 256;                       // 9216; 4 iters -> 36864 f4
    #pragma unroll
    for (int i = 0; i < 4; ++i) {
        int q = base + i * stride;
        f4 v = src[q];
        __builtin_nontemporal_store(v * gate, dst + q);
    }
}

extern "C" void kernel_launch(void* const* d_in, const int* in_sizes, int n_in,
                              void* d_out, int out_size, void* d_ws, size_t ws_size,
                              hipStream_t stream) {
    const float* prop  = (const float*)d_in[0];   // (B,P,4)
    const float* gcls  = (const float*)d_in[1];   // (B,R,C)
    const float* gtb   = (const float*)d_in[2];   // (B,R,4)
    const float* gmask = (const float*)d_in[3];   // (B,R,H,W)
    const int*   negI  = (const int*)d_in[4];     // (768,)
    const int*   posI  = (const int*)d_in[5];     // (256,)
    float* out   = (float*)d_out;
    float* posfW = (float*)d_ws;                  // 2048 floats of scratch

    iou_pos_kernel<<<8, 256, 0, stream>>>(prop, gtb, posfW);
    rois_bbox_kernel<<<8, 256, 0, stream>>>(prop, gtb, negI, posI, posfW, out);
    cls_wmma_kernel<<<B_ * 64, 32, 0, stream>>>(gcls, negI, posI, posfW, out);
    mask_kernel<<<dim3(36, B_ * NPOS), 256, 0, stream>>>(gmask, posI, posfW, out);
}